// get_model_84610855731465
// MI455X (gfx1250) — compile-verified
//
#include <hip/hip_runtime.h>
#include <cfloat>

typedef __attribute__((ext_vector_type(2))) float v2f;
typedef __attribute__((ext_vector_type(8))) float v8f;

__device__ __forceinline__ v8f wmma_f32_k4(v2f a, v2f b, v8f c) {
  // D = A(16x4) * B(4x16) + C(16x16), all f32.
  return __builtin_amdgcn_wmma_f32_16x16x4_f32(false, a, false, b, (short)0, c,
                                               false, false);
}

// gfx1250 async global->LDS copy (16B per lane), ASYNCcnt-tracked.
__device__ __forceinline__ void async_ld_b128(unsigned lds_off, const float* g) {
  asm volatile("global_load_async_to_lds_b128 %0, %1, off"
               :
               : "v"(lds_off), "v"(g)
               : "memory");
}

__device__ __forceinline__ void wait_async0() {
#if __has_builtin(__builtin_amdgcn_s_wait_asynccnt)
  __builtin_amdgcn_s_wait_asynccnt(0);
#else
  asm volatile("s_wait_asynccnt 0x0" ::: "memory");
#endif
}

__device__ __forceinline__ unsigned lds_off32(const void* p) {
  // generic pointer to LDS: low 32 bits hold the workgroup-relative offset
  return (unsigned)(size_t)p;
}

// ---------------------------------------------------------------- zero
__global__ void k_zero(float* p, int n) {
  int i = blockIdx.x * blockDim.x + threadIdx.x;
  if (i < n) p[i] = 0.0f;
}

// ---------------------------------------------------- per-instance stats
__global__ void k_stats_partial(const float* __restrict__ pc,
                                float* __restrict__ stats) {
  int b = blockIdx.x, tid = threadIdx.x;
  const float* p = pc + (size_t)b * 2048 * 3;
  float mn[3] = {FLT_MAX, FLT_MAX, FLT_MAX};
  float mx[3] = {-FLT_MAX, -FLT_MAX, -FLT_MAX};
  float sm[3] = {0.f, 0.f, 0.f};
  for (int i = tid; i < 2048; i += 256) {
#pragma unroll
    for (int c = 0; c < 3; ++c) {
      float v = p[i * 3 + c];
      mn[c] = fminf(mn[c], v);
      mx[c] = fmaxf(mx[c], v);
      sm[c] += v;
    }
  }
  __shared__ float s[256 * 9];
#pragma unroll
  for (int c = 0; c < 3; ++c) {
    s[tid * 9 + c] = mn[c];
    s[tid * 9 + 3 + c] = mx[c];
    s[tid * 9 + 6 + c] = sm[c];
  }
  __syncthreads();
  for (int st = 128; st > 0; st >>= 1) {
    if (tid < st) {
#pragma unroll
      for (int c = 0; c < 3; ++c) {
        s[tid * 9 + c] = fminf(s[tid * 9 + c], s[(tid + st) * 9 + c]);
        s[tid * 9 + 3 + c] = fmaxf(s[tid * 9 + 3 + c], s[(tid + st) * 9 + 3 + c]);
        s[tid * 9 + 6 + c] += s[(tid + st) * 9 + 6 + c];
      }
    }
    __syncthreads();
  }
  if (tid == 0) {
#pragma unroll
    for (int k = 0; k < 9; ++k) stats[b * 9 + k] = s[k];
  }
}

__global__ void k_stats_final(const float* __restrict__ stats,
                              float* __restrict__ normp,
                              float* __restrict__ centers) {
  int tid = threadIdx.x;
  __shared__ float s[128 * 6];
  __shared__ float osc[4];
#pragma unroll
  for (int c = 0; c < 3; ++c) {
    s[tid * 6 + c] = stats[tid * 9 + c];
    s[tid * 6 + 3 + c] = stats[tid * 9 + 3 + c];
  }
  __syncthreads();
  for (int st = 64; st > 0; st >>= 1) {
    if (tid < st) {
#pragma unroll
      for (int c = 0; c < 3; ++c) {
        s[tid * 6 + c] = fminf(s[tid * 6 + c], s[(tid + st) * 6 + c]);
        s[tid * 6 + 3 + c] = fmaxf(s[tid * 6 + 3 + c], s[(tid + st) * 6 + 3 + c]);
      }
    }
    __syncthreads();
  }
  if (tid == 0) {
    float sc = -FLT_MAX;
#pragma unroll
    for (int c = 0; c < 3; ++c) {
      float off = (s[3 + c] + s[c]) * 0.5f;
      osc[c] = off;
      normp[c] = off;
      sc = fmaxf(sc, s[3 + c] - s[c]);
    }
    osc[3] = sc;
    normp[3] = sc;
  }
  __syncthreads();
  float inv = 1.0f / osc[3];
#pragma unroll
  for (int c = 0; c < 3; ++c)
    centers[tid * 3 + c] =
        (stats[tid * 9 + 6 + c] * (1.0f / 2048.0f) - osc[c]) * inv;
}

// ------------------------------------------------------------------ FPS
// Invariant under the uniform normalization -> run on raw coords.
__global__ void k_fps(const float* __restrict__ pc, float* __restrict__ samp) {
  int b = blockIdx.x, tid = threadIdx.x;
  __shared__ float X[2048], Y[2048], Z[2048];
  __shared__ float rv[8];
  __shared__ int ri[8];
  __shared__ float bestv;
  __shared__ int besti;
  const float* p = pc + (size_t)b * 2048 * 3;
  for (int i = tid; i < 2048; i += 256) {
    X[i] = p[i * 3 + 0];
    Y[i] = p[i * 3 + 1];
    Z[i] = p[i * 3 + 2];
  }
  __syncthreads();
  float px[8], py[8], pz[8], d[8];
#pragma unroll
  for (int u = 0; u < 8; ++u) {
    int i = tid * 8 + u;
    px[u] = X[i]; py[u] = Y[i]; pz[u] = Z[i];
    d[u] = FLT_MAX;
  }
  int last = 0;
  if (tid == 0) {
    float* o = samp + (size_t)b * 768;
    o[0] = X[0]; o[1] = Y[0]; o[2] = Z[0];
  }
  int lane = tid & 31, w = tid >> 5;
  for (int it = 1; it < 256; ++it) {
    float lx = X[last], ly = Y[last], lz = Z[last];
    float bv = -FLT_MAX;
    int bi = 0;
#pragma unroll
    for (int u = 0; u < 8; ++u) {
      float dx = px[u] - lx, dy = py[u] - ly, dz = pz[u] - lz;
      float dd = dx * dx + dy * dy + dz * dz;
      d[u] = fminf(d[u], dd);
      if (d[u] > bv) { bv = d[u]; bi = tid * 8 + u; }
    }
    for (int m = 16; m >= 1; m >>= 1) {
      float ov = __shfl_xor(bv, m);
      int oi = __shfl_xor(bi, m);
      if (ov > bv || (ov == bv && oi < bi)) { bv = ov; bi = oi; }
    }
    if (lane == 0) { rv[w] = bv; ri[w] = bi; }
    __syncthreads();
    if (w == 0) {
      float v2 = (lane < 8) ? rv[lane] : -FLT_MAX;
      int i2 = (lane < 8) ? ri[lane] : 0x7fffffff;
      for (int m = 4; m >= 1; m >>= 1) {
        float ov = __shfl_xor(v2, m);
        int oi = __shfl_xor(i2, m);
        if (ov > v2 || (ov == v2 && oi < i2)) { v2 = ov; i2 = oi; }
      }
      if (lane == 0) { bestv = v2; besti = i2; }
    }
    __syncthreads();
    last = besti;
    if (tid == 0) {
      float* o = samp + (size_t)b * 768 + it * 3;
      o[0] = X[last]; o[1] = Y[last]; o[2] = Z[last];
    }
  }
  (void)bestv;
}

// --------------------------------------------------- fused PointNet + max
__global__ void k_pointnet(const float* __restrict__ pts, int inst_stride3,
                           int chunks_per_blk, const float* __restrict__ normp,
                           const float* __restrict__ W1, const float* __restrict__ b1,
                           const float* __restrict__ W2, const float* __restrict__ b2,
                           const float* __restrict__ W3, const float* __restrict__ b3,
                           float* __restrict__ codes, int colOff) {
  int b = blockIdx.x, split = blockIdx.y, tid = threadIdx.x;
  __shared__ float H1[64][68];
  __shared__ float H2[64][132];
  __shared__ unsigned accm[512];
  __shared__ float P[64 * 3];
  for (int n = tid; n < 512; n += 256) accm[n] = 0u;
  float off0 = normp[0], off1 = normp[1], off2 = normp[2];
  float inv = 1.0f / normp[3];
  int lane = tid & 31, w = tid >> 5;
  int half = lane >> 4, r = lane & 15;
  const float* base = pts + (size_t)b * inst_stride3;
  int c0 = split * chunks_per_blk;
  for (int ch = 0; ch < chunks_per_blk; ++ch) {
    int p0 = (c0 + ch) * 64;
    __syncthreads();
    for (int i = tid; i < 64; i += 256) {
      const float* q = base + (size_t)(p0 + i) * 3;
      P[i * 3 + 0] = (q[0] - off0) * inv;
      P[i * 3 + 1] = (q[1] - off1) * inv;
      P[i * 3 + 2] = (q[2] - off2) * inv;
    }
    __syncthreads();
    for (int e = tid; e < 4096; e += 256) {
      int qi = e >> 6, o = e & 63;
      float v = P[qi * 3 + 0] * W1[o] + P[qi * 3 + 1] * W1[64 + o] +
                P[qi * 3 + 2] * W1[128 + o] + b1[o];
      H1[qi][o] = v > 0.f ? v : 0.f;
    }
    __syncthreads();
    for (int t = w * 4; t < w * 4 + 4; ++t) {
      int m0 = (t >> 3) * 16, n0 = (t & 7) * 16;
      v8f acc = {0.f, 0.f, 0.f, 0.f, 0.f, 0.f, 0.f, 0.f};
      for (int k = 0; k < 64; k += 4) {
        int ka = k + 2 * half;
        v2f a; a.x = H1[m0 + r][ka]; a.y = H1[m0 + r][ka + 1];
        v2f bb; bb.x = W2[ka * 128 + n0 + r]; bb.y = W2[(ka + 1) * 128 + n0 + r];
        acc = wmma_f32_k4(a, bb, acc);
      }
      float bias = b2[n0 + r];
#pragma unroll
      for (int v = 0; v < 8; ++v) {
        float val = acc[v] + bias;
        H2[m0 + v + 8 * half][n0 + r] = val > 0.f ? val : 0.f;
      }
    }
    __syncthreads();
    for (int t = w * 16; t < w * 16 + 16; ++t) {
      int m0 = (t >> 5) * 16, n0 = (t & 31) * 16;
      v8f acc = {0.f, 0.f, 0.f, 0.f, 0.f, 0.f, 0.f, 0.f};
      for (int k = 0; k < 128; k += 4) {
        int ka = k + 2 * half;
        v2f a; a.x = H2[m0 + r][ka]; a.y = H2[m0 + r][ka + 1];
        v2f bb; bb.x = W3[ka * 512 + n0 + r]; bb.y = W3[(ka + 1) * 512 + n0 + r];
        acc = wmma_f32_k4(a, bb, acc);
      }
      float bias = b3[n0 + r];
      float mx = 0.f;
#pragma unroll
      for (int v = 0; v < 8; ++v) {
        float val = acc[v] + bias;
        mx = fmaxf(mx, val > 0.f ? val : 0.f);
      }
      mx = fmaxf(mx, __shfl_xor(mx, 16));
      if (half == 0) atomicMax(&accm[n0 + r], __float_as_uint(mx));
    }
  }
  __syncthreads();
  unsigned* gout = (unsigned*)codes + (size_t)b * 1536 + colOff;
  for (int n = tid; n < 512; n += 256) atomicMax(&gout[n], accm[n]);
}

// -------------------------------------------- LDS-tiled WMMA GEMM (async)
// C[M,N] = act(A[M,K] @ W[K,N] + bias). Block tile 128x128, 8 waves, each
// wave owns a 16x128 strip (8 accumulators). K staged in 16-wide blocks,
// double-buffered in LDS via global_load_async_to_lds_b128 (ASYNCcnt).
// Requires: M%128==0, N%128==0, K%16==0, ldc==N.
// diffmode: A-row p is feats[i]-feats[j] (i=p/127, j skips i) staged in VALU.
__global__ void k_gemm_tiled(const float* __restrict__ A,
                             const float* __restrict__ W,
                             const float* __restrict__ bias,
                             float* __restrict__ C, int M, int K, int N,
                             int relu, int diffmode) {
  __shared__ __align__(16) float As[2][128][20];  // 16 cols + 4 pad
  __shared__ __align__(16) float Bs[2][16][132];  // 128 cols + 4 pad
  int tid = threadIdx.x;
  int lane = tid & 31, w = tid >> 5;
  int half = lane >> 4, r = lane & 15;
  int mblk = blockIdx.x, nblk = blockIdx.y;
  int nkb = K >> 4;

  auto stage = [&](int kb, int buf) {
    int k0 = kb << 4;
    // A tile: 128 rows x 16 cols = 512 x float4
    if (diffmode) {
#pragma unroll
      for (int q = 0; q < 2; ++q) {
        int v = tid + q * 256;
        int rowl = v >> 2;
        int c = (v & 3) * 4;
        int p = mblk * 128 + rowl;
        int i = p / 127;
        int jj = p % 127;
        int j = jj + (jj >= i ? 1 : 0);
        const float4 va = *(const float4*)(A + (size_t)i * K + k0 + c);
        const float4 vb = *(const float4*)(A + (size_t)j * K + k0 + c);
        float4 dv;
        dv.x = va.x - vb.x; dv.y = va.y - vb.y;
        dv.z = va.z - vb.z; dv.w = va.w - vb.w;
        *(float4*)&As[buf][rowl][c] = dv;
      }
    } else {
#pragma unroll
      for (int q = 0; q < 2; ++q) {
        int v = tid + q * 256;
        int rowl = v >> 2;
        int c = (v & 3) * 4;
        async_ld_b128(lds_off32(&As[buf][rowl][c]),
                      A + (size_t)(mblk * 128 + rowl) * K + k0 + c);
      }
    }
    // B tile: 16 rows x 128 cols = 512 x float4
#pragma unroll
    for (int q = 0; q < 2; ++q) {
      int v = tid + q * 256;
      int rowl = v >> 5;
      int c = (v & 31) * 4;
      async_ld_b128(lds_off32(&Bs[buf][rowl][c]),
                    W + (size_t)(k0 + rowl) * N + nblk * 128 + c);
    }
  };

  v8f acc[8];
#pragma unroll
  for (int ni = 0; ni < 8; ++ni)
    acc[ni] = (v8f){0.f, 0.f, 0.f, 0.f, 0.f, 0.f, 0.f, 0.f};

  stage(0, 0);
  for (int kb = 0; kb < nkb; ++kb) {
    int buf = kb & 1;
    wait_async0();
    __syncthreads();
    if (kb + 1 < nkb) stage(kb + 1, buf ^ 1);
#pragma unroll
    for (int ks = 0; ks < 4; ++ks) {
      int k = ks * 4;
      int ka = k + 2 * half;
      v2f a;
      a.x = As[buf][w * 16 + r][ka];
      a.y = As[buf][w * 16 + r][ka + 1];
#pragma unroll
      for (int ni = 0; ni < 8; ++ni) {
        v2f bb;
        bb.x = Bs[buf][ka][ni * 16 + r];
        bb.y = Bs[buf][ka + 1][ni * 16 + r];
        acc[ni] = wmma_f32_k4(a, bb, acc[ni]);
      }
    }
    __syncthreads();
  }

#pragma unroll
  for (int ni = 0; ni < 8; ++ni) {
    int col = nblk * 128 + ni * 16 + r;
    float bv = bias[col];
#pragma unroll
    for (int v = 0; v < 8; ++v) {
      float val = acc[ni][v] + bv;
      if (relu) val = val > 0.f ? val : 0.f;
      int row = mblk * 128 + w * 16 + v + 8 * half;
      C[(size_t)row * N + col] = val;
    }
  }
}

// --------------------------- direct WMMA GEMM for the tiny padded head
__global__ void k_gemm_small(const float* __restrict__ A,
                             const float* __restrict__ W,
                             const float* __restrict__ bias,
                             float* __restrict__ C, int M, int K, int N,
                             int ldc, int nstore, int relu) {
  int tid = threadIdx.x;
  int lane = tid & 31, w = tid >> 5;
  int half = lane >> 4, r = lane & 15;
  int ntiles = N >> 4;
  int tiles = (M >> 4) * ntiles;
  int t = blockIdx.x * 8 + w;
  if (t >= tiles) return;
  int m0 = (t / ntiles) * 16, n0 = (t % ntiles) * 16;
  const float* arow = A + (size_t)(m0 + r) * K;
  v8f acc = {0.f, 0.f, 0.f, 0.f, 0.f, 0.f, 0.f, 0.f};
  for (int k = 0; k < K; k += 4) {
    int ka = k + 2 * half;
    v2f a; a.x = arow[ka]; a.y = arow[ka + 1];
    v2f bb;
    bb.x = W[(size_t)ka * N + n0 + r];
    bb.y = W[(size_t)(ka + 1) * N + n0 + r];
    acc = wmma_f32_k4(a, bb, acc);
  }
  int col = n0 + r;
  float bv = bias[col];
#pragma unroll
  for (int v = 0; v < 8; ++v) {
    float val = acc[v] + bv;
    if (relu) val = val > 0.f ? val : 0.f;
    int orow = m0 + v + 8 * half;
    if (col < nstore) C[(size_t)orow * ldc + col] = val;
  }
}

// ------------------------------------------------------------ small utils
__global__ void k_feats(const float* __restrict__ pcc,
                        const float* __restrict__ centers,
                        float* __restrict__ feats) {
  int i = blockIdx.x * 256 + threadIdx.x;
  if (i >= 128 * 528) return;
  int b = i / 528, n = i % 528;
  float v;
  if (n < 512) v = pcc[b * 512 + n];
  else if (n < 515) v = centers[b * 3 + (n - 512)];
  else v = 0.f;
  feats[i] = v;
}

__global__ void k_padw515(const float* __restrict__ W, float* __restrict__ Wp) {
  int i = blockIdx.x * 256 + threadIdx.x;
  if (i >= 528 * 512) return;
  int kk = i / 512, n = i % 512;
  Wp[i] = (kk < 515) ? W[kk * 512 + n] : 0.f;
}

__global__ void k_padcls(const float* __restrict__ W, const float* __restrict__ b,
                         float* __restrict__ Wp, float* __restrict__ bp) {
  int i = blockIdx.x * 256 + threadIdx.x;
  if (i < 128 * 32) {
    int kk = i >> 5, n = i & 31;
    Wp[i] = (n < 27) ? W[kk * 27 + n] : 0.f;
  } else if (i < 128 * 32 + 32) {
    int n = i - 128 * 32;
    bp[n] = (n < 27) ? b[n] : 0.f;
  }
}

// ---------------------------------------------------------------- launch
extern "C" void kernel_launch(void* const* d_in, const int* in_sizes, int n_in,
                              void* d_out, int out_size, void* d_ws,
                              size_t ws_size, hipStream_t stream) {
  (void)in_sizes; (void)n_in; (void)out_size; (void)ws_size;
  const float* pc = (const float*)d_in[0];
  const float* pnW[3][3];
  const float* pnB[3][3];
  for (int p = 0; p < 3; ++p)
    for (int l = 0; l < 3; ++l) {
      pnW[p][l] = (const float*)d_in[1 + p * 6 + l * 2];
      pnB[p][l] = (const float*)d_in[1 + p * 6 + l * 2 + 1];
    }
  const float* m1W1 = (const float*)d_in[19]; const float* m1b1 = (const float*)d_in[20];
  const float* m1W2 = (const float*)d_in[21]; const float* m1b2 = (const float*)d_in[22];
  const float* m1W3 = (const float*)d_in[23]; const float* m1b3 = (const float*)d_in[24];
  const float* m2W1 = (const float*)d_in[25]; const float* m2b1 = (const float*)d_in[26];
  const float* m2W2 = (const float*)d_in[27]; const float* m2b2 = (const float*)d_in[28];
  const float* cW1 = (const float*)d_in[29]; const float* cb1 = (const float*)d_in[30];
  const float* cW2 = (const float*)d_in[31]; const float* cb2 = (const float*)d_in[32];
  const float* cW3 = (const float*)d_in[33]; const float* cb3 = (const float*)d_in[34];
  float* out = (float*)d_out;

  float* ws = (float*)d_ws;
  size_t o = 0;
  float* stats   = ws + o; o += 128 * 9;
  float* normp   = ws + o; o += 4;
  float* centers = ws + o; o += 128 * 3;
  float* samp    = ws + o; o += 128 * 256 * 3;
  float* codes   = ws + o; o += 128 * 1536;
  float* h1      = ws + o; o += 128 * 1024;
  float* h2      = ws + o; o += 128 * 1024;
  float* pcc     = ws + o; o += 128 * 512;
  float* feats   = ws + o; o += 128 * 528;
  float* Wp      = ws + o; o += 528 * 512;
  float* m2h     = ws + o; o += 16256 * 512;
  float* ch1     = ws + o; o += 16256 * 256;
  float* ch2     = ws + o; o += 16256 * 128;
  float* clsWp   = ws + o; o += 128 * 32;
  float* clsbp   = ws + o; o += 32;

  float* pred_codes = out + 16256 * 27;  // second output region

  auto gemm = [&](const float* A, const float* W, const float* bias, float* C,
                  int M, int K, int N, int relu, int diff) {
    k_gemm_tiled<<<dim3(M / 128, N / 128), 256, 0, stream>>>(A, W, bias, C, M,
                                                             K, N, relu, diff);
  };

  k_zero<<<(128 * 1536 + 255) / 256, 256, 0, stream>>>(codes, 128 * 1536);
  k_stats_partial<<<128, 256, 0, stream>>>(pc, stats);
  k_stats_final<<<1, 128, 0, stream>>>(stats, normp, centers);
  k_fps<<<128, 256, 0, stream>>>(pc, samp);

  // PointNets: pc (2048 pts, 8 splits x 4 chunks), pc_256 (2 x 2), pc_128 (1 x 2)
  k_pointnet<<<dim3(128, 8), 256, 0, stream>>>(pc, 2048 * 3, 4, normp,
      pnW[0][0], pnB[0][0], pnW[0][1], pnB[0][1], pnW[0][2], pnB[0][2], codes, 0);
  k_pointnet<<<dim3(128, 2), 256, 0, stream>>>(samp, 256 * 3, 2, normp,
      pnW[1][0], pnB[1][0], pnW[1][1], pnB[1][1], pnW[1][2], pnB[1][2], codes, 512);
  k_pointnet<<<dim3(128, 1), 256, 0, stream>>>(samp, 256 * 3, 2, normp,
      pnW[2][0], pnB[2][0], pnW[2][1], pnB[2][1], pnW[2][2], pnB[2][2], codes, 1024);

  // mlp1: 1536 -> 1024 -> 1024 -> 512 (relu)
  gemm(codes, m1W1, m1b1, h1, 128, 1536, 1024, 1, 0);
  gemm(h1, m1W2, m1b2, h2, 128, 1024, 1024, 1, 0);
  gemm(h2, m1W3, m1b3, pcc, 128, 1024, 512, 1, 0);

  // feats[128,528] = [pc_codes | centers | 0-pad]; Wp = mlp2.W1 -> 528 rows
  k_feats<<<(128 * 528 + 255) / 256, 256, 0, stream>>>(pcc, centers, feats);
  k_padw515<<<(528 * 512 + 255) / 256, 256, 0, stream>>>(m2W1, Wp);

  // mlp2 over all 16256 ordered pairs (A-rows = feats[i]-feats[j] on the fly)
  gemm(feats, Wp, m2b1, m2h, 16256, 528, 512, 1, 1);
  gemm(m2h, m2W2, m2b2, pred_codes, 16256, 512, 512, 1, 0);

  // classifier: 512 -> 256 -> 128 -> 27 (head padded to N=32)
  gemm(pred_codes, cW1, cb1, ch1, 16256, 512, 256, 1, 0);
  gemm(ch1, cW2, cb2, ch2, 16256, 256, 128, 1, 0);
  k_padcls<<<(128 * 32 + 32 + 255) / 256, 256, 0, stream>>>(cW3, cb3, clsWp, clsbp);
  {
    int tiles = (16256 / 16) * (32 / 16);
    k_gemm_small<<<(tiles + 7) / 8, 256, 0, stream>>>(ch2, clsWp, clsbp, out,
                                                      16256, 128, 32, 27, 27, 0);
  }
}